// Uni_75076028334696
// MI455X (gfx1250) — compile-verified
//
#include <hip/hip_runtime.h>
#include <stdint.h>

#define DF 16
#define NODES_PER_BLOCK 64
#define EDGE_CAP 4096
#define N_LAYERS 12

typedef unsigned int u32;
typedef __attribute__((ext_vector_type(4))) u32 u32x4;
typedef __attribute__((ext_vector_type(8))) u32 u32x8;
typedef __attribute__((ext_vector_type(4))) float f32x4;

// ---- pass 1: in-degree count (dst only, matching reference) + non-isolated flags
__global__ __launch_bounds__(256) void count_kernel(const int* __restrict__ src,
                                                    const int* __restrict__ dst,
                                                    int* __restrict__ indeg,
                                                    int* __restrict__ flags, int E) {
  int e = blockIdx.x * 256 + threadIdx.x;
  if (e < E) {
    int s = src[e], d = dst[e];
    atomicAdd(&indeg[d], 1);
    flags[s] = 1;
    flags[d] = 1;
  }
}

// ---- pass 2: exclusive scan of indeg -> row_ptr (single block, 1024 threads)
__global__ __launch_bounds__(1024) void scan_kernel(const int* __restrict__ indeg,
                                                    int* __restrict__ row_ptr, int n) {
  __shared__ int part[1024];
  int t = threadIdx.x;
  int chunk = (n + 1023) >> 10;
  int beg = t * chunk;
  int end = beg + chunk; if (end > n) end = n;
  int sum = 0;
  for (int i = beg; i < end; ++i) sum += indeg[i];
  part[t] = sum;
  __syncthreads();
  for (int off = 1; off < 1024; off <<= 1) {
    int v = (t >= off) ? part[t - off] : 0;
    __syncthreads();
    part[t] += v;
    __syncthreads();
  }
  int prefix = (t == 0) ? 0 : part[t - 1];
  for (int i = beg; i < end; ++i) { row_ptr[i] = prefix; prefix += indeg[i]; }
  if (t == 1023) row_ptr[n] = part[1023];
}

// ---- pass 3: invs = rsqrt(deg) and pre-scaled state g0 = invs * (flags ? x : 0)
// One thread per feature-quad (float4).
__global__ __launch_bounds__(256) void init_kernel(const float* __restrict__ x_,
                                                   const int* __restrict__ indeg,
                                                   const int* __restrict__ flags,
                                                   float* __restrict__ invs,
                                                   float* __restrict__ g0_, int n) {
  int t = blockIdx.x * 256 + threadIdx.x;
  if (t < n * 4) {
    int i = t >> 2;
    float isv = rsqrtf((float)(indeg[i] + 1));
    if ((t & 3) == 0) invs[i] = isv;
    const f32x4* x4 = (const f32x4*)x_;
    f32x4 z = {0.0f, 0.0f, 0.0f, 0.0f};
    ((f32x4*)g0_)[t] = flags[i] ? (x4[t] * isv) : z;
  }
}

// ---- pass 4: CSR fill (sources bucketed by dst)
__global__ __launch_bounds__(256) void fill_kernel(const int* __restrict__ src,
                                                   const int* __restrict__ dst,
                                                   const int* __restrict__ row_ptr,
                                                   int* __restrict__ fillc,
                                                   int* __restrict__ src_sorted, int E) {
  int e = blockIdx.x * 256 + threadIdx.x;
  if (e < E) {
    int d = dst[e];
    int pos = row_ptr[d] + atomicAdd(&fillc[d], 1);
    src_sorted[pos] = src[e];
  }
}

// ---- pass 5: sort each row so per-call FP summation order is bit-deterministic
__global__ __launch_bounds__(256) void sort_rows_kernel(const int* __restrict__ row_ptr,
                                                        int* __restrict__ a, int n) {
  int i = blockIdx.x * 256 + threadIdx.x;
  if (i < n) {
    int rb = row_ptr[i], re = row_ptr[i + 1];
    for (int k = rb + 1; k < re; ++k) {
      int v = a[k];
      int j = k - 1;
      while (j >= rb && a[j] > v) { a[j + 1] = a[j]; --j; }
      a[j + 1] = v;
    }
  }
}

// ---- propagation layer in the pre-scaled domain: zero atomics, pure pull.
//   last==0:  gout[i] = invs[i]^2 * (g[i] + sum g[src])      (next state g)
//   last==1:  gout[i] = invs[i]   * (g[i] + sum g[src])      (final h)
// 4 lanes per node (lane-quad = float4 of features) -> each edge gather is one
// coalesced 64B line via 4x global_load_b128. The block's contiguous CSR slice
// is bulk-prefetched into LDS by the Tensor Data Mover (async-tensor path).
__global__ __launch_bounds__(256) void propagate_kernel(const float* __restrict__ gin_,
                                                        float* __restrict__ gout_,
                                                        const float* __restrict__ invs,
                                                        const int* __restrict__ row_ptr,
                                                        const int* __restrict__ src_sorted,
                                                        int n, int last) {
  const f32x4* gin = (const f32x4*)gin_;
  f32x4* gout = (f32x4*)gout_;
  __shared__ int s_src[EDGE_CAP];
  int first = blockIdx.x * NODES_PER_BLOCK;
  int lastE = first + NODES_PER_BLOCK; if (lastE > n) lastE = n;
  int e0 = row_ptr[first];
  int e1 = row_ptr[lastE];
  int L = e1 - e0;
  bool useLds = (L > 0) && (L <= EDGE_CAP);

  if (threadIdx.x < 32) {  // wave 0 issues the TDM op (per-wave instruction)
    if (useLds) {
      u32 ldsoff = (u32)(uintptr_t)(&s_src[0]);            // low 32 bits = LDS offset
      uint64_t ga = (uint64_t)(uintptr_t)(src_sorted + e0);
      u32 ga_lo = __builtin_amdgcn_readfirstlane((u32)ga);
      u32 ga_hi = __builtin_amdgcn_readfirstlane((u32)(ga >> 32));
      u32 Lr    = __builtin_amdgcn_readfirstlane((u32)L);
      // D# group0: count=1 | lds_addr | global_addr[56:0] | type=2 ("image")
      u32x4 g0;
      g0.x = 1u;
      g0.y = ldsoff;
      g0.z = ga_lo;
      g0.w = (ga_hi & 0x1FFFFFFu) | (2u << 30);
      // D# group1: data_size=4B; tensor_dim0=L, tensor_dim1=1; tile_dim0=L, tile_dim1=1
      u32x8 g1;
      g1.s0 = 0x00020000u;                          // workgroup_mask=0, data_size=2 (4B)
      g1.s1 = (Lr & 0xFFFFu) << 16;                 // tensor_dim0[15:0]
      g1.s2 = 0x00010000u | ((Lr >> 16) & 0xFFFFu); // tensor_dim0[31:16], tensor_dim1=1
      g1.s3 = (Lr & 0xFFFFu) << 16;                 // tile_dim0 = L
      g1.s4 = 1u;                                   // tile_dim1 = 1, tile_dim2 unused
      g1.s5 = Lr;                                   // tensor_dim0_stride = L
      g1.s6 = 0u;
      g1.s7 = 0u;
      asm volatile("tensor_load_to_lds %0, %1" :: "s"(g0), "s"(g1) : "memory");
      __builtin_amdgcn_s_wait_tensorcnt(0);
    }
  }
  __syncthreads();

  int local = threadIdx.x >> 2;
  int q = threadIdx.x & 3;
  int i = first + local;
  if (i >= n) return;

  float isv = invs[i];
  int rb = row_ptr[i], re = row_ptr[i + 1];
  f32x4 acc0 = gin[i * 4 + q];   // self term g[i] (scaling applied at the end)
  f32x4 acc1 = {0.0f, 0.0f, 0.0f, 0.0f};
  if (useLds) {
    int k = rb;
    for (; k + 1 < re; k += 2) {              // 2 accumulators for latency hiding
      int s0 = s_src[k - e0];
      int s1 = s_src[k + 1 - e0];
      acc0 += gin[s0 * 4 + q];
      acc1 += gin[s1 * 4 + q];
    }
    if (k < re) acc0 += gin[s_src[k - e0] * 4 + q];
  } else {                                    // fallback (practically never taken)
    for (int k = rb; k < re; ++k) acc0 += gin[src_sorted[k] * 4 + q];
  }
  float scale = last ? isv : (isv * isv);
  gout[i * 4 + q] = (acc0 + acc1) * scale;
}

extern "C" void kernel_launch(void* const* d_in, const int* in_sizes, int n_in,
                              void* d_out, int out_size, void* d_ws, size_t ws_size,
                              hipStream_t stream) {
  const float* x = (const float*)d_in[0];   // [N, 16, 1] float32
  const int* ei = (const int*)d_in[1];      // [2, E] int32
  int n = in_sizes[0] / DF;
  int E = in_sizes[1] / 2;
  const int* src = ei;
  const int* dst = ei + E;

  // workspace carve-up (~27 MB total)
  char* w = (char*)d_ws;
  auto carve = [&](size_t bytes) -> char* {
    char* p = w;
    w += (bytes + 255) & ~(size_t)255;
    return p;
  };
  int* indeg       = (int*)carve((size_t)n * 4);
  int* flags       = (int*)carve((size_t)n * 4);
  int* fillc       = (int*)carve((size_t)n * 4);
  float* invs      = (float*)carve((size_t)n * 4);
  int* row_ptr     = (int*)carve((size_t)(n + 1) * 4);
  int* src_sorted  = (int*)carve((size_t)E * 4);
  float* gA        = (float*)carve((size_t)n * DF * 4);
  float* gB        = (float*)carve((size_t)n * DF * 4);
  (void)ws_size; (void)n_in; (void)out_size;

  hipMemsetAsync(indeg, 0, (size_t)n * 4, stream);
  hipMemsetAsync(flags, 0, (size_t)n * 4, stream);
  hipMemsetAsync(fillc, 0, (size_t)n * 4, stream);

  int eb = (E + 255) / 256;
  count_kernel<<<eb, 256, 0, stream>>>(src, dst, indeg, flags, E);
  scan_kernel<<<1, 1024, 0, stream>>>(indeg, row_ptr, n);
  init_kernel<<<(n * 4 + 255) / 256, 256, 0, stream>>>(x, indeg, flags, invs, gA, n);
  fill_kernel<<<eb, 256, 0, stream>>>(src, dst, row_ptr, fillc, src_sorted, E);
  sort_rows_kernel<<<(n + 255) / 256, 256, 0, stream>>>(row_ptr, src_sorted, n);

  int pb = (n + NODES_PER_BLOCK - 1) / NODES_PER_BLOCK;
  float* bufs[2] = {gA, gB};
  for (int l = 0; l < N_LAYERS; ++l) {
    const float* gi = bufs[l & 1];
    int last = (l == N_LAYERS - 1);
    float* go = last ? (float*)d_out : bufs[(l + 1) & 1];
    propagate_kernel<<<pb, 256, 0, stream>>>(gi, go, invs, row_ptr, src_sorted, n, last);
  }
}